// Qwen3MoEGate_69715909149521
// MI455X (gfx1250) — compile-verified
//
#include <hip/hip_runtime.h>
#include <hip/hip_bf16.h>

typedef float v2f __attribute__((ext_vector_type(2)));
typedef float v8f __attribute__((ext_vector_type(8)));

#define HID   2048   // hidden size (K)
#define NEXP  64     // experts (N)
#define TOPK  8
#define KC    512    // K-chunk staged in LDS
#define LDS_STRIDE (KC + 4)   // pad to avoid stride-512 bank conflicts

// ---------------------------------------------------------------------------
// Kernel 1: logits = hidden @ W_gate^T via V_WMMA_F32_16X16X4_F32
//   grid.x = ntok/16 blocks, 128 threads (4 waves). Wave w computes the
//   16-token x 16-expert tile [m0:m0+16, 16w:16w+16], K-accumulated in
//   steps of 4 (fp32 WMMA shape).
// ---------------------------------------------------------------------------
__global__ __launch_bounds__(128)
void moe_gate_logits_wmma(const float* __restrict__ hidden,
                          const float* __restrict__ Wg,
                          float* __restrict__ logits)
{
    __shared__ float As[16 * LDS_STRIDE];   // 16 rows x KC (+pad) = ~33 KB

    const int tid  = threadIdx.x;
    const int wave = tid >> 5;
    const int lane = tid & 31;
    const int m0   = blockIdx.x * 16;       // token tile base
    const int e0   = wave * 16;             // expert tile base
    const int half = lane >> 4;             // 0 -> K {0,1}, 1 -> K {2,3}
    const int l15  = lane & 15;             // M row (A) / N col (B)

    v8f c = {};  // 16x16 f32 accumulator tile (8 VGPRs)

    // B fragment source row: W_gate[e0 + l15][.] (row-major [64][2048]).
    const float* wrow = Wg + (size_t)(e0 + l15) * HID + half * 2;

    for (int kc = 0; kc < HID; kc += KC) {
        // --- cooperative staged load of the A tile chunk (coalesced b128) ---
        for (int i = tid; i < 16 * (KC / 4); i += 128) {
            const int row = i >> 7;          // KC/4 == 128 float4 per row
            const int c4  = i & 127;
            const float* src = hidden + (size_t)(m0 + row) * HID + kc + c4 * 4;
            *reinterpret_cast<float4*>(&As[row * LDS_STRIDE + c4 * 4]) =
                *reinterpret_cast<const float4*>(src);
            if (kc + KC < HID)  // prefetch next K-chunk of this row into caches
                __builtin_prefetch(src + KC, 0, 0);
        }
        __syncthreads();

        // --- WMMA accumulation over this K-chunk ---
        const float* arow = As + l15 * LDS_STRIDE + half * 2;
#pragma unroll 8
        for (int kk = 0; kk < KC; kk += 4) {
            v2f a = *reinterpret_cast<const v2f*>(arow + kk);            // LDS b64
            v2f b = *reinterpret_cast<const v2f*>(wrow + kc + kk);       // L2 b64
            c = __builtin_amdgcn_wmma_f32_16x16x4_f32(
                    /*neg_a=*/false, a, /*neg_b=*/false, b,
                    /*c_mod=*/(short)0, c, /*reuse_a=*/false, /*reuse_b=*/false);
        }
        __syncthreads();
    }

    // D layout: VGPR r holds (row = r + 8*half, col = l15) of the 16x16 tile.
#pragma unroll
    for (int r = 0; r < 8; ++r) {
        logits[(size_t)(m0 + r + 8 * half) * NEXP + (e0 + l15)] = c[r];
    }
}

// ---------------------------------------------------------------------------
// Kernel 2: per-token softmax over 64 experts + top-8 + renormalize.
//   One thread per token; 64 logits live in registers. Renormalized top-k
//   softmax == exp(sel)/sum(exp(sel)), so no full-sum division is needed.
// ---------------------------------------------------------------------------
__global__ __launch_bounds__(256)
void moe_gate_softmax_topk(const float* __restrict__ logits,
                           float* __restrict__ sel_w,
                           int*   __restrict__ sel_idx,
                           int ntok)
{
    const int t = blockIdx.x * blockDim.x + threadIdx.x;
    if (t >= ntok) return;

    float p[NEXP];
    const float4* lp = reinterpret_cast<const float4*>(logits + (size_t)t * NEXP);

    float mx = -INFINITY;
#pragma unroll
    for (int i = 0; i < NEXP / 4; ++i) {
        float4 v = lp[i];
        p[i * 4 + 0] = v.x; p[i * 4 + 1] = v.y;
        p[i * 4 + 2] = v.z; p[i * 4 + 3] = v.w;
        mx = fmaxf(mx, fmaxf(fmaxf(v.x, v.y), fmaxf(v.z, v.w)));
    }
#pragma unroll
    for (int i = 0; i < NEXP; ++i)
        p[i] = __expf(p[i] - mx);   // softmax numerators (shared denom cancels)

    float wsel[TOPK];
    int   isel[TOPK];
    float ssum = 0.0f;
#pragma unroll
    for (int j = 0; j < TOPK; ++j) {
        float best = -1.0f;
        int   bi   = 0;
#pragma unroll
        for (int i = 0; i < NEXP; ++i) {
            if (p[i] > best) { best = p[i]; bi = i; }   // strict > : lowest index wins ties
        }
        wsel[j] = best;
        isel[j] = bi;
        ssum   += best;
        p[bi]   = -1.0f;
    }
    const float inv = 1.0f / ssum;
#pragma unroll
    for (int j = 0; j < TOPK; ++j) {
        sel_w[(size_t)t * TOPK + j]   = wsel[j] * inv;
        sel_idx[(size_t)t * TOPK + j] = isel[j];
    }
}

// ---------------------------------------------------------------------------
extern "C" void kernel_launch(void* const* d_in, const int* in_sizes, int n_in,
                              void* d_out, int out_size, void* d_ws, size_t ws_size,
                              hipStream_t stream)
{
    const float* hidden = (const float*)d_in[0];   // [4,4096,2048] f32
    const float* Wg     = (const float*)d_in[1];   // [64,2048] f32
    // d_in[2] = top_k scalar (== 8, baked in as TOPK)

    const int ntok = in_sizes[0] / HID;            // 16384 tokens

    float* out     = (float*)d_out;
    float* sel_w   = out;                                   // [ntok,8] f32
    int*   sel_idx = (int*)(out + (size_t)ntok * TOPK);     // [ntok,8] i32 bits
    float* logits  = out + (size_t)ntok * TOPK * 2;         // [ntok,64] f32

    moe_gate_logits_wmma<<<ntok / 16, 128, 0, stream>>>(hidden, Wg, logits);
    moe_gate_softmax_topk<<<(ntok + 255) / 256, 256, 0, stream>>>(
        logits, sel_w, sel_idx, ntok);
}